// attention_49933289783869
// MI455X (gfx1250) — compile-verified
//
#include <hip/hip_runtime.h>
#include <math.h>

typedef __attribute__((ext_vector_type(2))) float v2f;
typedef __attribute__((ext_vector_type(8))) float v8f;

#define BDIM   64
#define SDIM   1024
#define HDIM   1000
#define CHUNK  64                 // timesteps per partial-softmax block
#define NCHUNK (SDIM / CHUNK)     // 16 chunks per batch

// ---------------------------------------------------------------------------
// K1: q[b,n] = sum_k h_tgt[b,k] * W[k,n]    (bias dropped: softmax-invariant)
// Full-precision WMMA: V_WMMA_F32_16X16X4_F32, one wave per 16x16 C tile.
// A fragment (16x4 f32, 2 VGPRs): lanes 0-15 hold M=lane K={0,1}; lanes 16-31
// hold M=lane-16 K={2,3}.  B fragment (4x16) mirrors it.  D: VGPR r -> rows
// m0+r (lanes 0-15) and m0+r+8 (lanes 16-31), col = n0 + (lane&15).
// ---------------------------------------------------------------------------
__global__ __launch_bounds__(32)
void qproj_wmma(const float* __restrict__ A, const float* __restrict__ Wm,
                float* __restrict__ q) {
  const int lane = threadIdx.x;
  const int n0   = blockIdx.x * 16;
  const int m0   = blockIdx.y * 16;
  const int half = lane >> 4;        // 0 or 1
  const int l15  = lane & 15;
  const int kb   = half * 2;         // K sub-offset: 0 or 2
  const int arow = m0 + l15;
  const int bcol = n0 + l15;
  const float bmask = (bcol < HDIM) ? 1.0f : 0.0f;  // pad last N tile with 0
  const int   bcolc = (bcol < HDIM) ? bcol : 0;

  const float* arow_p = A + (size_t)arow * HDIM;
  v8f c = {};
  for (int k = 0; k < HDIM; k += 4) {
    v2f a = *(const v2f*)(arow_p + k + kb);     // K=1000 is 8B-aligned per lane
    v2f b;
    b.x = bmask * Wm[(size_t)(k + kb)     * HDIM + bcolc];
    b.y = bmask * Wm[(size_t)(k + kb + 1) * HDIM + bcolc];
    c = __builtin_amdgcn_wmma_f32_16x16x4_f32(false, a, false, b,
                                              (short)0, c, false, false);
  }
  if (bcol < HDIM) {
    const int drow = m0 + half * 8;
#pragma unroll
    for (int r = 0; r < 8; ++r)
      q[(size_t)(drow + r) * HDIM + bcol] = c[r];
  }
}

// ---------------------------------------------------------------------------
// K2: flash-style partial pass. Block = (batch b, chunk of 64 timesteps).
// Streams its 64x1000 slice of h_src from HBM exactly once:
//   scores -> local max m, local exp-sum l, exp-weighted partial context.
// ---------------------------------------------------------------------------
__global__ __launch_bounds__(256)
void attn_partial(const float* __restrict__ hsrc, const float* __restrict__ q,
                  float* __restrict__ ml, float* __restrict__ cpart) {
  __shared__ float q_s[HDIM];
  __shared__ float s_s[CHUNK];   // raw scores
  __shared__ float p_s[CHUNK];   // exp(score - m)

  const int b     = blockIdx.y;
  const int chunk = blockIdx.x;
  const int t0    = chunk * CHUNK;
  const int tid   = threadIdx.x;
  const int lane  = tid & 31;
  const int wave  = tid >> 5;

  for (int h = tid; h < HDIM; h += 256) q_s[h] = q[(size_t)b * HDIM + h];
  __syncthreads();

  // scores: 8 waves x 8 rows each, lane-strided dot of length 1000
  for (int j = wave; j < CHUNK; j += 8) {
    const float* row = hsrc + ((size_t)(b * SDIM + t0 + j)) * HDIM;
    float part = 0.f;
    for (int h = lane; h < HDIM; h += 32) part += q_s[h] * row[h];
#pragma unroll
    for (int off = 16; off > 0; off >>= 1) part += __shfl_xor(part, off, 32);
    if (lane == 0) s_s[j] = part;
  }
  __syncthreads();

  // chunk-local softmax numerators (redundant per-thread scan of 64 values)
  float m = -INFINITY;
#pragma unroll
  for (int j = 0; j < CHUNK; ++j) m = fmaxf(m, s_s[j]);
  if (tid < CHUNK) p_s[tid] = __expf(s_s[tid] - m);
  __syncthreads();
  if (tid == 0) {
    float l = 0.f;
#pragma unroll
    for (int j = 0; j < CHUNK; ++j) l += p_s[j];
    ml[(size_t)(b * NCHUNK + chunk) * 2 + 0] = m;
    ml[(size_t)(b * NCHUNK + chunk) * 2 + 1] = l;
  }

  // partial context: each thread owns cols tid, +256, +512, (+768 if valid)
  const int c0 = tid;
  const bool v3 = (tid + 768) < HDIM;
  const int  c3 = v3 ? (tid + 768) : 0;
  float acc0 = 0.f, acc1 = 0.f, acc2 = 0.f, acc3 = 0.f;
  const float* base = hsrc + ((size_t)(b * SDIM + t0)) * HDIM;
#pragma unroll 4
  for (int j = 0; j < CHUNK; ++j) {
    const float pj  = p_s[j];
    const float* row = base + (size_t)j * HDIM;
    acc0 += pj * row[c0];
    acc1 += pj * row[c0 + 256];
    acc2 += pj * row[c0 + 512];
    acc3 += pj * row[c3];
  }
  float* cp = cpart + ((size_t)(b * NCHUNK + chunk)) * HDIM;
  cp[c0]       = acc0;
  cp[c0 + 256] = acc1;
  cp[c0 + 512] = acc2;
  if (v3) cp[tid + 768] = acc3;
}

// ---------------------------------------------------------------------------
// K3: merge 16 chunk partials per batch:
//   M = max m_i;  L = sum l_i e^{m_i-M};  out = (sum e^{m_i-M} c_i) / L
// ---------------------------------------------------------------------------
__global__ __launch_bounds__(256)
void attn_combine(const float* __restrict__ ml, const float* __restrict__ cpart,
                  float* __restrict__ out) {
  const int b   = blockIdx.x;
  const int tid = threadIdx.x;

  float M = -INFINITY;
#pragma unroll
  for (int i = 0; i < NCHUNK; ++i)
    M = fmaxf(M, ml[(size_t)(b * NCHUNK + i) * 2]);
  float f[NCHUNK];
  float L = 0.f;
#pragma unroll
  for (int i = 0; i < NCHUNK; ++i) {
    const float fi = __expf(ml[(size_t)(b * NCHUNK + i) * 2] - M);
    L += fi * ml[(size_t)(b * NCHUNK + i) * 2 + 1];
    f[i] = fi;
  }
  const float invL = 1.0f / L;
  for (int h = tid; h < HDIM; h += 256) {
    float acc = 0.f;
#pragma unroll
    for (int i = 0; i < NCHUNK; ++i)
      acc += f[i] * cpart[((size_t)(b * NCHUNK + i)) * HDIM + h];
    out[(size_t)b * HDIM + h] = acc * invL;
  }
}

// ---------------------------------------------------------------------------
extern "C" void kernel_launch(void* const* d_in, const int* in_sizes, int n_in,
                              void* d_out, int out_size, void* d_ws, size_t ws_size,
                              hipStream_t stream) {
  const float* h_tgt = (const float*)d_in[0];   // [64,1,1000]
  const float* h_src = (const float*)d_in[1];   // [64,1024,1000]
  const float* W     = (const float*)d_in[2];   // [1000,1000]
  // d_in[3] = bias: dropped — softmax is shift-invariant per row.
  float* out = (float*)d_out;                   // [64,1,1000] f32

  char*  ws = (char*)d_ws;
  float* q  = (float*)ws;                                  // 64*1000 f32 (250 KB)
  float* ml = (float*)(ws + 256 * 1024);                   // 64*16*2 f32
  float* cp = (float*)(ws + 256 * 1024 + 16 * 1024);       // 64*16*1000 f32 (~4 MB)

  qproj_wmma  <<<dim3((HDIM + 15) / 16, BDIM / 16), 32,  0, stream>>>(h_tgt, W, q);
  attn_partial<<<dim3(NCHUNK, BDIM),               256, 0, stream>>>(h_src, q, ml, cp);
  attn_combine<<<BDIM,                             256, 0, stream>>>(ml, cp, out);
}